// Follow_loss_1580547966785
// MI455X (gfx1250) — compile-verified
//
#include <hip/hip_runtime.h>

typedef __attribute__((ext_vector_type(4))) float f4;
typedef __attribute__((ext_vector_type(2))) float v2f;
typedef __attribute__((ext_vector_type(8))) float v8f;

// Exact f32 sum of the 32 per-lane partials of a wave using one
// v_wmma_f32_16x16x4_f32.  A (16x4, layout per CDNA5 ISA 7.12.2):
//   VGPR0: lanes 0-15 -> A[m,0]=p[m], lanes 16-31 -> A[m,2]=p[m+16]
//   VGPR1: zeros (K=1 / K=3)
// B = all ones (4x16) -> layout irrelevant since every element is 1.0.
// D[m,n] = p[m] + p[m+16] for every n.  D VGPR r: lanes 0-15 hold M=r,
// lanes 16-31 hold M=r+8.  Hence sum(c0..c7) at lane 0 = sum(p[0..7],p[16..23])
// and at lane 16 = sum(p[8..15],p[24..31]); their sum is the full wave sum.
__device__ __forceinline__ float wave_sum32(float p) {
  v2f a; a.x = p;    a.y = 0.0f;
  v2f b; b.x = 1.0f; b.y = 1.0f;
  v8f c = {};
  c = __builtin_amdgcn_wmma_f32_16x16x4_f32(
      /*neg_a=*/false, a, /*neg_b=*/false, b,
      /*c_mod=*/(short)0, c, /*reuse_a=*/false, /*reuse_b=*/false);
  float s = ((c[0] + c[1]) + (c[2] + c[3])) + ((c[4] + c[5]) + (c[6] + c[7]));
  int i0  = __builtin_amdgcn_readlane(__float_as_int(s), 0);
  int i16 = __builtin_amdgcn_readlane(__float_as_int(s), 16);
  return __int_as_float(i0) + __int_as_float(i16);
}

// Pass 1: stream the three quaternion arrays (128-bit non-temporal loads),
// compute product -> normalize -> squared error, reduce to one float / block.
extern "C" __global__ void __launch_bounds__(256)
follow_loss_partial(const f4* __restrict__ vq, const f4* __restrict__ rq,
                    const f4* __restrict__ rp, float* __restrict__ bpart,
                    int n) {
  float acc = 0.0f;
  const int stride = (int)(gridDim.x * blockDim.x);
  for (int i = (int)(blockIdx.x * blockDim.x + threadIdx.x); i < n; i += stride) {
    const f4 q1 = __builtin_nontemporal_load(&rq[i]);
    const f4 q2 = __builtin_nontemporal_load(&rp[i]);
    const f4 v  = __builtin_nontemporal_load(&vq[i]);
    // quaternion product (x,y,z,w layout, q1 * q2), matching the reference
    const float x = q1.w * q2.x + q1.x * q2.w + q1.y * q2.z - q1.z * q2.y;
    const float y = q1.w * q2.y - q1.x * q2.z + q1.y * q2.w + q1.z * q2.x;
    const float z = q1.w * q2.z + q1.x * q2.y - q1.y * q2.x + q1.z * q2.w;
    const float w = q1.w * q2.w - q1.x * q2.x - q1.y * q2.y - q1.z * q2.z;
    const float nrm = sqrtf(x * x + y * y + z * z + w * w);
    float nx, ny, nz, nw;
    if (nrm > 1e-6f) {
      const float inv = 1.0f / nrm;  // one IEEE divide instead of four
      nx = x * inv; ny = y * inv; nz = z * inv; nw = w * inv;
    } else {
      nx = 0.0f; ny = 0.0f; nz = 0.0f; nw = w / w;  // NaN if w==0, like ref
    }
    const float dx = v.x - nx, dy = v.y - ny, dz = v.z - nz, dw = v.w - nw;
    acc += (dx * dx + dy * dy) + (dz * dz + dw * dw);
  }

  // wave32 reduce via WMMA (all 256 lanes active -> EXEC all ones)
  const float wsum = wave_sum32(acc);

  __shared__ float swave[8];
  const int lane = (int)(threadIdx.x & 31u);
  const int wid  = (int)(threadIdx.x >> 5u);
  if (lane == 0) swave[wid] = wsum;
  __syncthreads();
  if (threadIdx.x == 0) {
    float b = 0.0f;
#pragma unroll
    for (int k = 0; k < 8; ++k) b += swave[k];
    bpart[blockIdx.x] = b;
  }
}

// Pass 2: reduce block partials in f64, divide by element count, emit f32.
extern "C" __global__ void __launch_bounds__(256)
follow_loss_final(const float* __restrict__ bpart, int nb,
                  float* __restrict__ out, double inv_count) {
  __shared__ double sh[256];
  double s = 0.0;
  for (int i = (int)threadIdx.x; i < nb; i += 256) s += (double)bpart[i];
  sh[threadIdx.x] = s;
  __syncthreads();
  for (int off = 128; off > 0; off >>= 1) {
    if ((int)threadIdx.x < off) sh[threadIdx.x] += sh[threadIdx.x + off];
    __syncthreads();
  }
  if (threadIdx.x == 0) out[0] = (float)(sh[0] * inv_count);
}

extern "C" void kernel_launch(void* const* d_in, const int* in_sizes, int n_in,
                              void* d_out, int out_size, void* d_ws, size_t ws_size,
                              hipStream_t stream) {
  const f4* vq = (const f4*)d_in[0];   // virtual_quat  (B,4) f32
  const f4* rq = (const f4*)d_in[1];   // real_quat     (B,4) f32
  const f4* rp = (const f4*)d_in[2];   // real_postion  (B,4) f32
  float* out = (float*)d_out;

  const int n = in_sizes[0] / 4;       // number of quaternion rows

  int nb = 1024;                       // 1024 blocks * 8 waves = 8192 waves
  if (ws_size < (size_t)nb * sizeof(float)) {
    nb = (int)(ws_size / sizeof(float));
    if (nb < 1) nb = 1;
  }
  float* bpart = (float*)d_ws;

  follow_loss_partial<<<nb, 256, 0, stream>>>(vq, rq, rp, bpart, n);

  const double inv_count = 1.0 / (4.0 * (double)n);
  follow_loss_final<<<1, 256, 0, stream>>>(bpart, nb, out, inv_count);
}